// DeformableTransformer_73418170957878
// MI455X (gfx1250) — compile-verified
//
#include <hip/hip_runtime.h>

// DeformableTransformer warp: out[b,c,h,w] = bilinear(source[b,c], grid(h,w))
// Memory-bound gather (~260 MB min traffic @ 23.3 TB/s ~= 11 us floor). No
// matmul structure -> WMMA inapplicable; optimize data movement instead:
//   * async global->LDS staging of the displacement tile (ASYNCcnt path)
//   * non-temporal output stores so the 128 MB source stays in the 192 MB L2
//   * weights computed once per thread, reused across its 16 planes
//   * paired-tap b64 loads: x-adjacent taps fetched as one 8-byte load, with
//     border handling folded into a one-time weight swap (no per-plane selects)
//   * fully unrolled 16-plane inner loop for deep load-level parallelism

#define Bc 8
#define Cc 16
#define Hc 512
#define Wc 512
#define HWc (Hc * Wc)          // 262144 elements per plane
#define PLANES (Bc * Cc)       // 128
#define CHUNK 16               // planes per thread
#define NCHUNK (PLANES / CHUNK)

struct F2 { float lo, hi; };   // loaded via memcpy: backend may use b64 align-4

__global__ __launch_bounds__(256) void warp_bilinear_kernel(
    const float* __restrict__ src,   // [128, 512, 512]
    const float* __restrict__ disp,  // [2, 512, 512]
    float* __restrict__ out)         // [128, 512, 512]
{
    const int tid   = threadIdx.x;
    const int pix   = blockIdx.x * 256 + tid;   // pix = h*512 + w, contiguous per block
    const int chunk = blockIdx.y;               // which group of 16 planes

    __shared__ float sdx[256];
    __shared__ float sdy[256];

    // ---- Stage this block's displacement tile into LDS via the CDNA5 async
    // global->LDS path (tracked with ASYNCcnt). Each lane pulls its own dx/dy;
    // dy sits one plane (1 MiB, fits the signed 24-bit ioffset) past dx.
    {
        unsigned ldsx = (unsigned)(unsigned long long)&sdx[tid];
        unsigned ldsy = (unsigned)(unsigned long long)&sdy[tid];
        unsigned voff = (unsigned)(pix * 4);
        unsigned long long base = (unsigned long long)disp;
        asm volatile(
            "global_load_async_to_lds_b32 %0, %2, %3\n\t"
            "global_load_async_to_lds_b32 %1, %2, %3 offset:1048576\n\t"
            "s_wait_asynccnt 0"
            :
            : "v"(ldsx), "v"(ldsy), "v"(voff), "s"(base)
            : "memory");
    }
    const float dx = sdx[tid];
    const float dy = sdy[tid];

    // ---- Per-pixel sampling coordinates (identical for all 128 planes).
    const int w = pix & (Wc - 1);
    const int h = pix >> 9;
    const float STEP = 2.0f / 511.0f;           // avoid v_div sequence
    const float gx = -1.0f + (float)w * STEP + dx;
    const float gy = -1.0f + (float)h * STEP + dy;
    const float x = (gx + 1.0f) * 0.5f * 511.0f;
    const float y = (gy + 1.0f) * 0.5f * 511.0f;

    const float x0f = floorf(x), y0f = floorf(y);
    const float wx1 = x - x0f, wy1 = y - y0f;
    const float wx0 = 1.0f - wx1, wy0 = 1.0f - wy1;
    const int x0 = (int)x0f, y0 = (int)y0f;
    const int x1 = x0 + 1, y1 = y0 + 1;

    // zeros-padding: invalid taps get weight 0, indices clamped.
    const float vx0 = (x0 >= 0 && x0 < Wc) ? 1.0f : 0.0f;
    const float vx1 = (x1 >= 0 && x1 < Wc) ? 1.0f : 0.0f;
    const float vy0 = (y0 >= 0 && y0 < Hc) ? 1.0f : 0.0f;
    const float vy1 = (y1 >= 0 && y1 < Hc) ? 1.0f : 0.0f;
    const float w00 = wx0 * wy0 * vx0 * vy0;
    const float w01 = wx1 * wy0 * vx1 * vy0;
    const float w10 = wx0 * wy1 * vx0 * vy1;
    const float w11 = wx1 * wy1 * vx1 * vy1;

    // Paired-tap addressing: both x-taps of a row come from one 8B load at
    // column xb = clamp(x0, 0, W-2). When x0 is outside [0, W-2] the two
    // columns swap roles (and the out-of-range tap's weight is already 0),
    // so a single per-pixel weight swap makes the inner loop select-free.
    const int  xb  = min(max(x0, 0), Wc - 2);
    const bool sel = (x0 >= 0) && (x0 <= Wc - 2);
    const float wa0 = sel ? w00 : w01;   // weight for column xb,   row y0
    const float wb0 = sel ? w01 : w00;   // weight for column xb+1, row y0
    const float wa1 = sel ? w10 : w11;   // weight for column xb,   row y1
    const float wb1 = sel ? w11 : w10;   // weight for column xb+1, row y1

    const int y0c = min(max(y0, 0), Hc - 1);
    const int y1c = min(max(y1, 0), Hc - 1);
    const int o0 = y0c * Wc + xb;
    const int o1 = y1c * Wc + xb;

    // ---- Gather + blend over this thread's 16 planes. Source taps use the
    // default (temporal) policy so they stay hot in the 192 MB L2; the output
    // is streamed with non-temporal stores so it does not evict the source.
    const float* sp = src + (size_t)chunk * CHUNK * HWc;
    float*       op = out + (size_t)chunk * CHUNK * HWc + (size_t)pix;

#pragma unroll
    for (int p = 0; p < CHUNK; ++p) {
        const float* pl = sp + (size_t)p * HWc;
        F2 a, b;
        __builtin_memcpy(&a, pl + o0, sizeof(F2));   // row y0: taps xb, xb+1
        __builtin_memcpy(&b, pl + o1, sizeof(F2));   // row y1: taps xb, xb+1
        const float r = fmaf(wa0, a.lo,
                        fmaf(wb0, a.hi,
                        fmaf(wa1, b.lo, wb1 * b.hi)));
        __builtin_nontemporal_store(r, op + (size_t)p * HWc);
    }
}

extern "C" void kernel_launch(void* const* d_in, const int* in_sizes, int n_in,
                              void* d_out, int out_size, void* d_ws, size_t ws_size,
                              hipStream_t stream) {
    (void)in_sizes; (void)n_in; (void)out_size; (void)d_ws; (void)ws_size;
    const float* src  = (const float*)d_in[0];  // source       [8,16,512,512] f32
    const float* disp = (const float*)d_in[1];  // displacement [1,2,512,512]  f32
    float* out = (float*)d_out;                 // [8,16,512,512] f32

    dim3 grid(HWc / 256, NCHUNK, 1);            // 1024 pixel-blocks x 8 plane-chunks
    warp_bilinear_kernel<<<grid, 256, 0, stream>>>(src, disp, out);
}